// MiniBatchDiscrimination_8495445312057
// MI455X (gfx1250) — compile-verified
//
#include <hip/hip_runtime.h>

typedef __attribute__((ext_vector_type(2))) float v2f;
typedef __attribute__((ext_vector_type(8))) float v8f;

#define NN 256     // batch
#define AA 1024    // inner dim
#define BB 128     // feature groups
#define CC 16      // per-group dim
#define BCC 2048   // BB*CC

// ---------------------------------------------------------------------------
// Kernel 1: M[n, b*16+c] = sum_a x[n,a] * T2[a, b*16+c]
// One wave computes one 16x16 tile of M via V_WMMA_F32_16X16X4_F32.
// A-frag (16x4 f32, 2 VGPRs): lane l (m = l&15, h = l>>4):
//   v0 = A[m, k + 2h], v1 = A[m, k + 2h + 1]        (ISA 7.12.2, 32-bit A 16x4)
// B-frag (4x16 f32, 2 VGPRs): lane l (n = l&15, h = l>>4):
//   v0 = B[k + 2h, n], v1 = B[k + 2h + 1, n]
// C/D (16x16 f32, 8 VGPRs): d[r] = D[r + 8h, n]
// ---------------------------------------------------------------------------
__global__ __launch_bounds__(256) void mbd_gemm_wmma(
    const float* __restrict__ x,    // [NN, AA]
    const float* __restrict__ T2,   // [AA, BCC]
    float* __restrict__ M)          // [NN, BCC]
{
    const int lane = threadIdx.x & 31;
    const int wave = threadIdx.x >> 5;
    const int tile = blockIdx.x * 8 + wave;   // 0..2047
    const int rowTile = tile >> 7;            // 0..15  (NN/16)
    const int colTile = tile & 127;           // 0..127 (BCC/16)
    const int row0 = rowTile * 16;
    const int col0 = colTile * 16;
    const int lm = lane & 15;
    const int h  = lane >> 4;                 // lane-half selects K pair

    v8f acc = {};
    const float* xrow = x  + (size_t)(row0 + lm) * AA;  // A row for this lane
    const float* bcol = T2 + (size_t)(col0 + lm);       // B column base

#pragma unroll 4
    for (int k = 0; k < AA; k += 4) {
        v2f a;  // two consecutive f32 -> single b64 load
        a.x = xrow[k + 2 * h];
        a.y = xrow[k + 2 * h + 1];
        v2f b;
        b.x = bcol[(size_t)(k + 2 * h) * BCC];
        b.y = bcol[(size_t)(k + 2 * h + 1) * BCC];
        // (neg_a, A, neg_b, B, c_mod, C, reuse_a, reuse_b)
        acc = __builtin_amdgcn_wmma_f32_16x16x4_f32(
            false, a, false, b, (short)0, acc, false, false);
    }

    float* mout = M + (size_t)(row0 + 8 * h) * BCC + col0 + lm;
#pragma unroll
    for (int r = 0; r < 8; ++r)
        mout[(size_t)r * BCC] = acc[r];
}

// ---------------------------------------------------------------------------
// Kernel 2: one block per b (128 blocks). Stage M[:, b, :] (256x16 f32 =
// 16 KB) in LDS; thread t = row k. Inner j loop reads slab[j*16+c] uniformly
// across the wave -> LDS broadcast, no bank conflicts.
// out[k,b] = sum_j exp(-sum_c |M[k,b,c]-M[j,b,c]|) - (k==0 ? 1 : C[k,N-1,b])
// (for k==N-1 the j==N-1 term equals the self term 1.0, so this covers the
//  reference's slicing quirk for all three cases)
// ---------------------------------------------------------------------------
__global__ __launch_bounds__(256) void mbd_pairwise(
    const float* __restrict__ M,    // [NN, BCC]
    float* __restrict__ out)        // [NN, BB]
{
    __shared__ float slab[NN * CC];           // 16 KB
    const int b = blockIdx.x;
    const int t = threadIdx.x;                // 0..255

    // cooperative load: 16 floats per thread, 64-byte aligned -> b128 loads
    const float4* src = (const float4*)(M + (size_t)t * BCC + b * CC);
    float4* dst = (float4*)(slab + t * CC);
    dst[0] = src[0]; dst[1] = src[1]; dst[2] = src[2]; dst[3] = src[3];
    __syncthreads();

    float mk[CC];
#pragma unroll
    for (int c = 0; c < CC; ++c) mk[c] = slab[t * CC + c];

    float s = 0.0f, last = 0.0f;
    for (int j = 0; j < NN; ++j) {
        const float* mj = slab + j * CC;
        float d = 0.0f;
#pragma unroll
        for (int c = 0; c < CC; ++c) d += fabsf(mk[c] - mj[c]);
        float e = __expf(-d);
        s += e;
        last = e;                              // holds j == NN-1 after loop
    }
    out[(size_t)t * BB + b] = s - (t == 0 ? 1.0f : last);
}

// ---------------------------------------------------------------------------
extern "C" void kernel_launch(void* const* d_in, const int* in_sizes, int n_in,
                              void* d_out, int out_size, void* d_ws, size_t ws_size,
                              hipStream_t stream) {
    const float* x  = (const float*)d_in[0];   // [256, 1024]
    const float* T2 = (const float*)d_in[1];   // [1024, 128, 16] == [1024, 2048]
    float* M = (float*)d_ws;                   // [256, 2048] scratch (2 MB)

    mbd_gemm_wmma<<<256, 256, 0, stream>>>(x, T2, M);
    mbd_pairwise<<<128, 256, 0, stream>>>(M, (float*)d_out);
}